// Model2_81527069213051
// MI455X (gfx1250) — compile-verified
//
#include <hip/hip_runtime.h>

typedef float v2f __attribute__((ext_vector_type(2)));
typedef float v4f __attribute__((ext_vector_type(4)));
typedef float v8f __attribute__((ext_vector_type(8)));

#define BN 64
#define NK 16
#define NF 64
#define OUT_STRIDE 369952          // per-batch flat output length
#define PAR_STRIDE 8               // floats per (bn,nk) param record in ws

// Output offsets within one batch row
#define OFF_S1 0                   // 16 ch * 128*128
#define OFF_S2 262144              // 16 ch * 64*64
#define OFF_S3 327680              // 16 ch * 32*32
#define OFF_S4 344064              // (16+64) ch * 16*16
#define OFF_S5 364544              // (4+64) ch * 8*8
#define OFF_S6 368896              // (2+64) ch * 4*4

__device__ __forceinline__ float heat_eval(float muY, float muX,
                                           float L00, float L10, float L11,
                                           float y, float x) {
    float dy = y - muY + 1e-6f;
    float dx = x - muX + 1e-6f;
    float p0 = L00 * dy;
    float p1 = fmaf(L10, dy, L11 * dx);
    return 1.0f / fmaf(p0, p0, fmaf(p1, p1, 1.0f));
}

// ---------------------------------------------------------------------------
// Kernel 1: per-(bn,nk) moments over the 64x64 softmax map -> mu, L_inv params
// One 256-thread block per map (1024 blocks). Streaming non-temporal 128-bit
// reads + wave32 shuffle reduction + cross-wave LDS reduction.
// ---------------------------------------------------------------------------
__global__ void moments_kernel(const float* __restrict__ part_maps,
                               float* __restrict__ params) {
    const int map = blockIdx.x;                 // bn*NK + nk
    const v4f* pm4 = (const v4f*)(part_maps + (size_t)map * 4096);
    const int tid = threadIdx.x;                // 256 threads
    const float step = 2.0f / 63.0f;

    float sy = 0.f, sx = 0.f, syy = 0.f, syx = 0.f, sxx = 0.f;
    #pragma unroll
    for (int base = 0; base < 1024; base += 256) {
        v4f v = __builtin_nontemporal_load(pm4 + base + tid);
        int idx = (base + tid) * 4;
        #pragma unroll
        for (int u = 0; u < 4; ++u) {
            int q = idx + u;
            float p = v[u];
            float y = -1.0f + (float)(q >> 6) * step;
            float x = -1.0f + (float)(q & 63) * step;
            sy  = fmaf(p, y, sy);
            sx  = fmaf(p, x, sx);
            syy = fmaf(p * y, y, syy);
            syx = fmaf(p * y, x, syx);
            sxx = fmaf(p * x, x, sxx);
        }
    }
    // wave32 reduction
    #pragma unroll
    for (int off = 16; off > 0; off >>= 1) {
        sy  += __shfl_down(sy,  off, 32);
        sx  += __shfl_down(sx,  off, 32);
        syy += __shfl_down(syy, off, 32);
        syx += __shfl_down(syx, off, 32);
        sxx += __shfl_down(sxx, off, 32);
    }
    __shared__ float red[8][5];
    const int wave = tid >> 5, lane = tid & 31;
    if (lane == 0) {
        red[wave][0] = sy; red[wave][1] = sx;
        red[wave][2] = syy; red[wave][3] = syx; red[wave][4] = sxx;
    }
    __syncthreads();
    if (tid == 0) {
        float t0 = 0.f, t1 = 0.f, t2 = 0.f, t3 = 0.f, t4 = 0.f;
        #pragma unroll
        for (int w = 0; w < 8; ++w) {
            t0 += red[w][0]; t1 += red[w][1];
            t2 += red[w][2]; t3 += red[w][3]; t4 += red[w][4];
        }
        float muY = t0, muX = t1;
        float s00 = t2 - muY * muY;
        float s01 = t3 - muY * muX;
        float s11 = t4 - muX * muX;
        float a = sqrtf(s00 + 1e-12f);
        float b = s01 / (a + 1e-12f);
        float c = sqrtf(s11 - b * b + 1e-12f);
        float scal = 0.8f / (a * c + 1e-12f);
        float* pr = params + (size_t)map * PAR_STRIDE;
        pr[0] = muY; pr[1] = muX;
        pr[2] = scal * c;        // L00
        pr[3] = -scal * b;       // L10
        pr[4] = scal * a;        // L11
    }
}

// ---------------------------------------------------------------------------
// Kernel 2: heat fields for stages 1-3 (no fmap). Pure streaming write:
// 4 pixels per thread in one row -> one non-temporal 128-bit store.
// grid = (h*w/1024, NK, BN); per-block params are uniform -> scalar loads.
// ---------------------------------------------------------------------------
__global__ void heat_kernel(const float* __restrict__ params,
                            float* __restrict__ out,
                            int h, int w, int out_off) {
    const int ch = blockIdx.y, bn = blockIdx.z;
    const int p0 = (blockIdx.x * blockDim.x + threadIdx.x) * 4;
    const int hw = h * w;
    const float* pr = params + (size_t)(bn * NK + ch) * PAR_STRIDE;
    const float muY = pr[0], muX = pr[1], L00 = pr[2], L10 = pr[3], L11 = pr[4];
    const float sy = 2.0f / (float)(h - 1), sx = 2.0f / (float)(w - 1);

    const int i = p0 / w, j = p0 - i * w;       // w % 4 == 0: one row per quad
    const float y = -1.0f + (float)i * sy;
    const float dy = y - muY + 1e-6f;
    const float q0 = L00 * dy;                  // shared across the quad
    const float t1 = L10 * dy;
    const float q00 = fmaf(q0, q0, 1.0f);

    v4f r;
    #pragma unroll
    for (int u = 0; u < 4; ++u) {
        float x = -1.0f + (float)(j + u) * sx;
        float dx = x - muX + 1e-6f;
        float p1 = fmaf(L11, dx, t1);
        r[u] = 1.0f / fmaf(p1, p1, q00);
    }
    v4f* dst = (v4f*)(out + (size_t)bn * OUT_STRIDE + out_off +
                      (size_t)ch * hw + p0);
    __builtin_nontemporal_store(r, dst);        // global_store_b128, TH=NT
}

// ---------------------------------------------------------------------------
// Kernel 3: stages 4-6. One 128-thread block (4 waves) per bn.
// Heats computed into LDS, part_heat streamed out, fmap = feat^T @ hs_norm
// done with V_WMMA_F32_16X16X4_F32 (f32 in, f32 acc -> matches reference).
// Wave w owns M-tile w (16 of the 64 feature channels).
// ---------------------------------------------------------------------------
__global__ void tail_kernel(const float* __restrict__ params,
                            const float* __restrict__ features,
                            float* __restrict__ out) {
    __shared__ float sPrm[NK][5];
    __shared__ float sHeat[NK][256];   // 16 KB
    __shared__ float sHsn[12][256];    // 12 KB (rows reused/padded per stage)
    __shared__ float sFeatA[NF][16];   // A-matrix staging: A[m][k]
    __shared__ float sDen[256];

    const int tid = threadIdx.x;       // 128 threads
    const int bn = blockIdx.x;
    float* outB = out + (size_t)bn * OUT_STRIDE;
    const float* featB = features + (size_t)bn * NK * NF;

    const int wv = tid >> 5;           // wave id = M tile
    const int lane = tid & 31;
    const int half = lane >> 4;        // 0: lanes 0-15, 1: lanes 16-31
    const int l16 = lane & 15;

    for (int v = tid; v < NK * 5; v += 128)
        sPrm[v / 5][v % 5] = params[(size_t)(bn * NK + v / 5) * PAR_STRIDE + (v % 5)];
    __syncthreads();

    // ================= stage 4: 16x16, part_depth=16, slice k=4..15 =========
    {
        const float st = 2.0f / 15.0f;
        for (int v = tid; v < NK * 256; v += 128) {
            int ch = v >> 8, p = v & 255;
            float y = -1.0f + (float)(p >> 4) * st;
            float x = -1.0f + (float)(p & 15) * st;
            float hv = heat_eval(sPrm[ch][0], sPrm[ch][1], sPrm[ch][2],
                                 sPrm[ch][3], sPrm[ch][4], y, x);
            sHeat[ch][p] = hv;
            outB[OFF_S4 + v] = hv;                       // v == ch*256 + p
        }
        __syncthreads();
        for (int p = tid; p < 256; p += 128) {
            float s = 1.0f;
            #pragma unroll
            for (int k = 4; k < 16; ++k) s += sHeat[k][p];
            sDen[p] = 1.0f / s;
        }
        for (int v = tid; v < NF * 16; v += 128) {
            int m = v >> 4, k = v & 15;
            sFeatA[m][k] = (k < 12) ? featB[(size_t)(4 + k) * NF + m] : 0.0f;
        }
        __syncthreads();
        for (int v = tid; v < 12 * 256; v += 128) {
            int k = v >> 8, p = v & 255;
            sHsn[k][p] = sHeat[4 + k][p] * sDen[p];
        }
        __syncthreads();

        const int mt = wv;
        for (int nt = 0; nt < 16; ++nt) {
            v8f acc = {};
            #pragma unroll
            for (int kc = 0; kc < 12; kc += 4) {
                int ka = kc + (half ? 2 : 0);
                v2f a, b;
                a.x = sFeatA[mt * 16 + l16][ka];
                a.y = sFeatA[mt * 16 + l16][ka + 1];
                b.x = sHsn[ka][nt * 16 + l16];
                b.y = sHsn[ka + 1][nt * 16 + l16];
                acc = __builtin_amdgcn_wmma_f32_16x16x4_f32(
                    false, a, false, b, (short)0, acc, false, false);
            }
            #pragma unroll
            for (int r = 0; r < 8; ++r) {
                int m = mt * 16 + r + (half ? 8 : 0);
                outB[OFF_S4 + (size_t)(16 + m) * 256 + nt * 16 + l16] = acc[r];
            }
        }
        __syncthreads();
    }

    // ================= stage 5: 8x8, part_depth=4, slice k=2..3 =============
    {
        const float st = 2.0f / 7.0f;
        for (int v = tid; v < 4 * 64; v += 128) {
            int ch = v >> 6, p = v & 63;
            float y = -1.0f + (float)(p >> 3) * st;
            float x = -1.0f + (float)(p & 7) * st;
            float hv = heat_eval(sPrm[ch][0], sPrm[ch][1], sPrm[ch][2],
                                 sPrm[ch][3], sPrm[ch][4], y, x);
            sHeat[ch][p] = hv;
            outB[OFF_S5 + v] = hv;
        }
        __syncthreads();
        for (int p = tid; p < 64; p += 128) {
            float inv = 1.0f / (1.0f + sHeat[2][p] + sHeat[3][p]);
            sHsn[0][p] = sHeat[2][p] * inv;
            sHsn[1][p] = sHeat[3][p] * inv;
            sHsn[2][p] = 0.0f;
            sHsn[3][p] = 0.0f;
        }
        for (int v = tid; v < NF * 16; v += 128) {
            int m = v >> 4, k = v & 15;
            sFeatA[m][k] = (k < 2) ? featB[(size_t)(2 + k) * NF + m] : 0.0f;
        }
        __syncthreads();

        const int mt = wv;
        for (int nt = 0; nt < 4; ++nt) {
            v8f acc = {};
            int ka = half ? 2 : 0;                 // K=2 padded to 4 with zeros
            v2f a, b;
            a.x = sFeatA[mt * 16 + l16][ka];
            a.y = sFeatA[mt * 16 + l16][ka + 1];
            b.x = sHsn[ka][nt * 16 + l16];
            b.y = sHsn[ka + 1][nt * 16 + l16];
            acc = __builtin_amdgcn_wmma_f32_16x16x4_f32(
                false, a, false, b, (short)0, acc, false, false);
            #pragma unroll
            for (int r = 0; r < 8; ++r) {
                int m = mt * 16 + r + (half ? 8 : 0);
                outB[OFF_S5 + (size_t)(4 + m) * 64 + nt * 16 + l16] = acc[r];
            }
        }
        __syncthreads();
    }

    // ================= stage 6: 4x4, part_depth=2, slice k=0..1 =============
    {
        const float st = 2.0f / 3.0f;
        for (int v = tid; v < 2 * 16; v += 128) {
            int ch = v >> 4, p = v & 15;
            float y = -1.0f + (float)(p >> 2) * st;
            float x = -1.0f + (float)(p & 3) * st;
            float hv = heat_eval(sPrm[ch][0], sPrm[ch][1], sPrm[ch][2],
                                 sPrm[ch][3], sPrm[ch][4], y, x);
            sHeat[ch][p] = hv;
            outB[OFF_S6 + v] = hv;
        }
        __syncthreads();
        if (tid < 16) {
            int p = tid;
            float inv = 1.0f / (1.0f + sHeat[0][p] + sHeat[1][p]);
            sHsn[0][p] = sHeat[0][p] * inv;
            sHsn[1][p] = sHeat[1][p] * inv;
            sHsn[2][p] = 0.0f;
            sHsn[3][p] = 0.0f;
        }
        for (int v = tid; v < NF * 16; v += 128) {
            int m = v >> 4, k = v & 15;
            sFeatA[m][k] = (k < 2) ? featB[(size_t)k * NF + m] : 0.0f;
        }
        __syncthreads();

        const int mt = wv;
        v8f acc = {};
        int ka = half ? 2 : 0;
        v2f a, b;
        a.x = sFeatA[mt * 16 + l16][ka];
        a.y = sFeatA[mt * 16 + l16][ka + 1];
        b.x = sHsn[ka][l16];
        b.y = sHsn[ka + 1][l16];
        acc = __builtin_amdgcn_wmma_f32_16x16x4_f32(
            false, a, false, b, (short)0, acc, false, false);
        #pragma unroll
        for (int r = 0; r < 8; ++r) {
            int m = mt * 16 + r + (half ? 8 : 0);
            outB[OFF_S6 + (size_t)(2 + m) * 16 + l16] = acc[r];
        }
    }
}

// ---------------------------------------------------------------------------
extern "C" void kernel_launch(void* const* d_in, const int* in_sizes, int n_in,
                              void* d_out, int out_size, void* d_ws, size_t ws_size,
                              hipStream_t stream) {
    const float* part_maps = (const float*)d_in[0];   // (64,16,64,64) f32
    const float* features  = (const float*)d_in[1];   // (64,16,64) f32
    float* out = (float*)d_out;                        // (64, 369952) f32
    float* params = (float*)d_ws;                      // 1024 * 8 floats = 32 KB

    moments_kernel<<<dim3(BN * NK), dim3(256), 0, stream>>>(part_maps, params);

    heat_kernel<<<dim3((128 * 128) / 1024, NK, BN), dim3(256), 0, stream>>>(
        params, out, 128, 128, OFF_S1);
    heat_kernel<<<dim3((64 * 64) / 1024, NK, BN), dim3(256), 0, stream>>>(
        params, out, 64, 64, OFF_S2);
    heat_kernel<<<dim3((32 * 32) / 1024, NK, BN), dim3(256), 0, stream>>>(
        params, out, 32, 32, OFF_S3);

    tail_kernel<<<dim3(BN), dim3(128), 0, stream>>>(params, features, out);
}